// JKNet_40638980555142
// MI455X (gfx1250) — compile-verified
//
#include <hip/hip_runtime.h>
#include <math.h>

typedef float v2f __attribute__((ext_vector_type(2)));
typedef float v8f __attribute__((ext_vector_type(8)));

#define NODES    100000
#define EDGES    800000
#define DIM      128
#define NLAYERS  4
#define NCLASSES 40

// ---------------- degree / norm ----------------
__global__ void deg_init_kernel(float* __restrict__ deg, int n) {
    int i = blockIdx.x * blockDim.x + threadIdx.x;
    if (i < n) deg[i] = 1.0f;                 // self-loop contributes 1
}

__global__ void deg_accum_kernel(const int* __restrict__ dst, float* __restrict__ deg, int e) {
    int i = blockIdx.x * blockDim.x + threadIdx.x;
    if (i < e) unsafeAtomicAdd(&deg[dst[i]], 1.0f);
}

__global__ void dinv_kernel(const float* __restrict__ deg, float* __restrict__ dinv, int n) {
    int i = blockIdx.x * blockDim.x + threadIdx.x;
    if (i < n) dinv[i] = rsqrtf(deg[i]);      // deg >= 1 always
}

// ---------------- zero fill ----------------
__global__ void fill_zero4_kernel(float4* __restrict__ p, long n4) {
    long i = (long)blockIdx.x * blockDim.x + threadIdx.x;
    if (i < n4) p[i] = make_float4(0.f, 0.f, 0.f, 0.f);
}

// ---------------- GEMM: out[N,128] = h[N,128] @ W[128,128] via f32 WMMA ----------------
// Block = 256 threads = 8 waves. Block covers 128 rows; each wave computes a
// 16-row x 128-col strip with 8 f32 16x16 accumulators (K reduced in steps of 4).
//
// LDS layout for W is k-pair interleaved: sW[(k>>1)*DIM + n] = {W[k][n], W[k+1][n]}
// so every B fragment is one contiguous 8-byte ds_load_b64 straight into the
// even-aligned VGPR pair WMMA SRC1 requires (no register-shuffle movs).
__global__ __launch_bounds__(256) void gemm_hW_kernel(const float* __restrict__ h,
                                                      const float* __restrict__ W,
                                                      float* __restrict__ out, int nrows) {
    __shared__ v2f sW[(DIM / 2) * DIM];       // 64 KB: full W, k-pair packed
    int tid = threadIdx.x;
    for (int i = tid; i < (DIM / 2) * DIM; i += 256) {
        int kp = i >> 7;                      // k-pair index 0..63
        int n  = i & (DIM - 1);
        v2f p;
        p.x = W[(2 * kp) * DIM + n];
        p.y = W[(2 * kp + 1) * DIM + n];
        sW[i] = p;
    }
    __syncthreads();

    int wave = tid >> 5;                      // wave32
    int lane = tid & 31;
    int row0 = blockIdx.x * 128 + wave * 16;
    if (row0 >= nrows) return;                // wave-uniform exit (after barrier)

    int lh    = lane & 15;                    // M (A) / N (B) index within tile
    int half  = lane >> 4;                    // 0: K={0,1}, 1: K={2,3}
    int khalf = half << 1;

    v8f acc[8] = {};
    const float* hrow = h + (size_t)(row0 + lh) * DIM;

    for (int k = 0; k < DIM; k += 4) {
        // A fragment 16x4: lane holds {h[row][k+khalf], h[row][k+khalf+1]}
        v2f a = *(const v2f*)(hrow + k + khalf);
        const v2f* brow = &sW[((k >> 1) + half) * DIM + lh];
        #pragma unroll
        for (int t = 0; t < 8; ++t) {
            // B fragment 4x16: one b64 load = {W[k+khalf][n], W[k+khalf+1][n]}
            v2f b = brow[t * 16];
            acc[t] = __builtin_amdgcn_wmma_f32_16x16x4_f32(
                false, a, false, b, (short)0, acc[t], false, false);
        }
    }

    // C/D layout: VGPR j, lane L -> row = j + 8*(L>=16), col = L%16
    int rbase = row0 + (half << 3);
    #pragma unroll
    for (int t = 0; t < 8; ++t) {
        int col = t * 16 + lh;
        #pragma unroll
        for (int j = 0; j < 8; ++j) {
            out[(size_t)(rbase + j) * DIM + col] = acc[t][j];
        }
    }
}

// ---------------- edge scatter: agg[dst] += hw[src] * dinv[src]*dinv[dst] ----------------
// One wave per edge; each lane moves a float4 (32 lanes x 4 = 128 features).
__global__ __launch_bounds__(256) void scatter_edges_kernel(const float* __restrict__ hw,
                                                            const int* __restrict__ src,
                                                            const int* __restrict__ dst,
                                                            const float* __restrict__ dinv,
                                                            float* __restrict__ agg, int nedges) {
    int e = blockIdx.x * (blockDim.x >> 5) + (threadIdx.x >> 5);
    if (e >= nedges) return;
    int lane = threadIdx.x & 31;
    int s = src[e];
    int d = dst[e];
    float nrm = dinv[s] * dinv[d];
    float4 v = ((const float4*)(hw + (size_t)s * DIM))[lane];
    float* o = agg + (size_t)d * DIM + lane * 4;
    unsafeAtomicAdd(o + 0, v.x * nrm);
    unsafeAtomicAdd(o + 1, v.y * nrm);
    unsafeAtomicAdd(o + 2, v.z * nrm);
    unsafeAtomicAdd(o + 3, v.w * nrm);
}

// ---------------- bias + self-loop + relu + JK max ----------------
__global__ void post_kernel(const float* __restrict__ agg, const float* __restrict__ hw,
                            const float* __restrict__ dinv, const float* __restrict__ b,
                            float* __restrict__ hout, float* __restrict__ jk,
                            int layer, long total) {
    long i = (long)blockIdx.x * blockDim.x + threadIdx.x;
    if (i >= total) return;
    int node = (int)(i >> 7);
    int d    = (int)(i & (DIM - 1));
    float di = dinv[node];
    float v  = agg[i] + hw[i] * di * di + b[d];   // self-loop folded in here
    v = fmaxf(v, 0.0f);
    hout[i] = v;
    jk[i] = (layer == 0) ? v : fmaxf(jk[i], v);
}

// ---------------- classifier + log_softmax: one block (128 thr) per node ----------------
__global__ __launch_bounds__(128) void classify_kernel(const float* __restrict__ jk,
                                                       const float* __restrict__ lin_w,
                                                       const float* __restrict__ lin_b,
                                                       float* __restrict__ out, int n) {
    __shared__ float sh[DIM];
    __shared__ float slog[NCLASSES];
    __shared__ float slse;
    int node = blockIdx.x;
    int t = threadIdx.x;
    sh[t] = jk[(size_t)node * DIM + t];
    __syncthreads();
    if (t < NCLASSES) {
        float acc = lin_b[t];
        #pragma unroll 8
        for (int d = 0; d < DIM; ++d) acc += sh[d] * lin_w[d * NCLASSES + t];
        slog[t] = acc;
    }
    __syncthreads();
    if (t == 0) {
        float m = slog[0];
        for (int c = 1; c < NCLASSES; ++c) m = fmaxf(m, slog[c]);
        float s = 0.0f;
        for (int c = 0; c < NCLASSES; ++c) s += __expf(slog[c] - m);
        slse = m + __logf(s);
    }
    __syncthreads();
    if (t < NCLASSES) out[(size_t)node * NCLASSES + t] = slog[t] - slse;
}

// ---------------- host-side orchestration ----------------
extern "C" void kernel_launch(void* const* d_in, const int* in_sizes, int n_in,
                              void* d_out, int out_size, void* d_ws, size_t ws_size,
                              hipStream_t stream) {
    const float* x     = (const float*)d_in[0];
    const int*   ei    = (const int*)d_in[1];   // [2, E]
    const float* Ws    = (const float*)d_in[2]; // [NL, D, D]
    const float* bs    = (const float*)d_in[3]; // [NL, D]
    const float* lin_w = (const float*)d_in[4]; // [D, NC]
    const float* lin_b = (const float*)d_in[5]; // [NC]
    float* out = (float*)d_out;

    const int* srcIdx = ei;
    const int* dstIdx = ei + EDGES;

    size_t nd = (size_t)NODES * DIM;
    float* ws   = (float*)d_ws;
    float* h    = ws;            // [N,D] current features
    float* hw   = h   + nd;      // [N,D] h @ W
    float* agg  = hw  + nd;      // [N,D] scatter accumulator
    float* jk   = agg + nd;      // [N,D] jumping-knowledge max
    float* deg  = jk  + nd;      // [N]
    float* dinv = deg + NODES;   // [N]

    // symmetric normalization terms
    deg_init_kernel <<<(NODES + 255) / 256, 256, 0, stream>>>(deg, NODES);
    deg_accum_kernel<<<(EDGES + 255) / 256, 256, 0, stream>>>(dstIdx, deg, EDGES);
    dinv_kernel     <<<(NODES + 255) / 256, 256, 0, stream>>>(deg, dinv, NODES);

    for (int l = 0; l < NLAYERS; ++l) {
        const float* hin = (l == 0) ? x : h;
        gemm_hW_kernel<<<(NODES + 127) / 128, 256, 0, stream>>>(
            hin, Ws + (size_t)l * DIM * DIM, hw, NODES);
        fill_zero4_kernel<<<(int)((nd / 4 + 255) / 256), 256, 0, stream>>>(
            (float4*)agg, (long)(nd / 4));
        scatter_edges_kernel<<<(EDGES + 7) / 8, 256, 0, stream>>>(
            hw, srcIdx, dstIdx, dinv, agg, EDGES);
        post_kernel<<<(int)((nd + 255) / 256), 256, 0, stream>>>(
            agg, hw, dinv, bs + (size_t)l * DIM, h, jk, l, (long)nd);
    }

    classify_kernel<<<NODES, 128, 0, stream>>>(jk, lin_w, lin_b, out, NODES);
}